// Kongming_SPMM_33208687133425
// MI455X (gfx1250) — compile-verified
//
#include <hip/hip_runtime.h>
#include <hip/hip_bf16.h>
#include <cstdint>

#define N_NODES 100000
#define N_RULES 20000
#define D 64          // feature width (f32)
#define TILE 256      // edges staged per async tile (3 KB of LDS per buffer)

// ---------------------------------------------------------------------------
// CDNA5 async helpers (ASYNCcnt-tracked fire-and-forget global -> LDS DMA)
// GV addressing form: "global_load_async_to_lds_bNN vdst(lds addr), vaddr64, off"
// ---------------------------------------------------------------------------
__device__ __forceinline__ void async_g2lds_b32(void* lds_dst, const void* gsrc) {
    uint32_t lds_addr = (uint32_t)(uintptr_t)lds_dst;   // low 32 bits of generic
                                                        // shared ptr = LDS offset
    asm volatile("global_load_async_to_lds_b32 %0, %1, off"
                 :
                 : "v"(lds_addr), "v"((uint64_t)(uintptr_t)gsrc)
                 : "memory");
}

__device__ __forceinline__ void async_g2lds_b128(void* lds_dst, const void* gsrc) {
    uint32_t lds_addr = (uint32_t)(uintptr_t)lds_dst;   // must be 16B aligned
    asm volatile("global_load_async_to_lds_b128 %0, %1, off"
                 :
                 : "v"(lds_addr), "v"((uint64_t)(uintptr_t)gsrc)
                 : "memory");
}

__device__ __forceinline__ void wait_async0() {
#if __has_builtin(__builtin_amdgcn_s_wait_asynccnt)
    __builtin_amdgcn_s_wait_asynccnt(0);
#else
    asm volatile("s_wait_asynccnt 0x0" ::: "memory");
#endif
}

// ---------------------------------------------------------------------------
// Zero-fill (float4 vectorized). n4 = number of float4 elements.
// ---------------------------------------------------------------------------
__global__ __launch_bounds__(256) void Kongming_zero4(float4* __restrict__ p, int n4) {
    int i = blockIdx.x * blockDim.x + threadIdx.x;
    if (i < n4) p[i] = make_float4(0.f, 0.f, 0.f, 0.f);
}

// ---------------------------------------------------------------------------
// COO scatter-SpMM:  out[rows[e]] += vals[e] * x[cols[e]]   (row width D=64 f32)
//
// Block = 256 threads = 16 edge-groups of 16 lanes; each lane owns one float4
// chunk of the 256B row. COO triples are staged in 256-edge tiles into LDS,
// double-buffered, via GLOBAL_LOAD_ASYNC_TO_LDS_B128 (192 x 16B issues per
// tile; wave-uniform array assignment, zero divergence): the async DMA fetches
// tile i+1 while tile i is consumed from LDS with broadcast ds_load_b32. The
// VMEM pipe is left for the two bounding operations: the b128 row gathers and
// the non-returning global_atomic_add_f32 scatters (both L2-resident: total
// footprint ~90 MB < 192 MB L2). Once a tile's cols are in LDS, each thread
// prefetches both 128B lines of its edge's source row (global_prefetch_b8),
// overlapping a full tile of gather latency.
// ---------------------------------------------------------------------------
__global__ __launch_bounds__(256) void Kongming_coo_scatter(
    const int*   __restrict__ rows,
    const int*   __restrict__ cols,
    const float* __restrict__ vals,
    const float* __restrict__ x,
    float*       __restrict__ out,
    int nnz)
{
    __shared__ alignas(16) int   sR[2][TILE];
    __shared__ alignas(16) int   sC[2][TILE];
    __shared__ alignas(16) float sV[2][TILE];

    const int t    = threadIdx.x;
    const int lane = t & 15;     // float4 chunk 0..15
    const int g    = t >> 4;     // edge-group 0..15 within block

    const long long tstride = (long long)gridDim.x * TILE;

    // Stage one tile: 192 threads each move one 16B chunk (4 edges) of one
    // array. arr is wave-uniform (waves 0-1: rows, 2-3: cols, 4-5: vals).
    auto stage = [&](int buf, long long base) {
        if (t < 192) {
            const int arr = t >> 6;           // 0=rows, 1=cols, 2=vals
            const int q   = (t & 63) * 4;     // element offset within tile
            const long long e = base + q;
            const void* gsrc;
            void*       ldst;
            if (arr == 0)      { gsrc = rows + e; ldst = &sR[buf][q]; }
            else if (arr == 1) { gsrc = cols + e; ldst = &sC[buf][q]; }
            else               { gsrc = vals + e; ldst = &sV[buf][q]; }
            if (e + 3 < (long long)nnz) {
                async_g2lds_b128(ldst, gsrc);
            } else {
                #pragma unroll
                for (int k = 0; k < 4; ++k)
                    if (e + k < (long long)nnz)
                        async_g2lds_b32((char*)ldst + 4 * k,
                                        (const char*)gsrc + 4 * k);
            }
        }
    };

    const long long base0 = (long long)blockIdx.x * TILE;
    int cur = 0;
    if (base0 < (long long)nnz) stage(0, base0);

    for (long long base = base0; base < (long long)nnz; base += tstride) {
        // My wave's async copies are complete; barrier makes every wave's
        // LDS writes visible block-wide.
        wait_async0();
        __syncthreads();

        // Kick off the DMA for the next tile into the other buffer.
        const long long nbase = base + tstride;
        if (nbase < (long long)nnz) stage(cur ^ 1, nbase);

        // Prefetch the full tile's gather rows (2 cachelines per 256B row).
        if (base + t < (long long)nnz) {
            const int cpf = sC[cur][t];
            __builtin_prefetch(x + (size_t)cpf * D,      0, 3);
            __builtin_prefetch(x + (size_t)cpf * D + 32, 0, 3);
        }

        // Consume: each 16-lane group handles 16 edges of this tile.
        #pragma unroll 4
        for (int p = 0; p < 16; ++p) {
            const int l = p * 16 + g;
            if (base + l < (long long)nnz) {
                const int   c = sC[cur][l];   // broadcast ds_load (conflict-free)
                const int   r = sR[cur][l];
                const float v = sV[cur][l];

                const float4 xv = *reinterpret_cast<const float4*>(
                    x + (size_t)c * D + (size_t)lane * 4);

                float* dst = out + (size_t)r * D + (size_t)lane * 4;
                atomicAdd(dst + 0, v * xv.x);
                atomicAdd(dst + 1, v * xv.y);
                atomicAdd(dst + 2, v * xv.z);
                atomicAdd(dst + 3, v * xv.w);
            }
        }
        cur ^= 1;
        // No trailing barrier needed: a buffer is only re-staged after the
        // next iteration's barrier, which every wave reaches only after it
        // finished consuming that buffer.
    }
}

// ---------------------------------------------------------------------------
// Launch helpers
// ---------------------------------------------------------------------------
static inline int scatter_blocks(int nnz) {
    long long tiles = ((long long)nnz + TILE - 1) / TILE;
    const long long cap = 2048;   // 16K waves in flight; >1 tile/block on big stages
    return (int)(tiles < cap ? tiles : cap);
}

extern "C" void kernel_launch(void* const* d_in, const int* in_sizes, int n_in,
                              void* d_out, int out_size, void* d_ws, size_t ws_size,
                              hipStream_t stream) {
    // Inputs, in setup_inputs() dict order:
    const float* x        = (const float*)d_in[0];   // (N_NODES, 64) f32
    const int*   v2v_rows = (const int*)  d_in[1];
    const int*   v2v_cols = (const int*)  d_in[2];
    const float* v2v_vals = (const float*)d_in[3];
    const int*   r2v_rows = (const int*)  d_in[4];
    const int*   r2v_cols = (const int*)  d_in[5];
    const float* r2v_vals = (const float*)d_in[6];
    const int*   r2r_rows = (const int*)  d_in[7];   // (2, nnz_r2r)
    const int*   r2r_cols = (const int*)  d_in[8];
    const float* r2r_vals = (const float*)d_in[9];
    const int*   v2r_rows = (const int*)  d_in[10];
    const int*   v2r_cols = (const int*)  d_in[11];
    const float* v2r_vals = (const float*)d_in[12];

    const int nnz_v2v = in_sizes[1];
    const int nnz_r2v = in_sizes[4];
    const int step    = 2;                       // fixed in reference
    const int nnz_r2r = in_sizes[7] / step;
    const int nnz_v2r = in_sizes[10];

    float* out = (float*)d_out;                  // (N_NODES, 64) f32

    // Workspace layout: rule accumulator + snapshot buffer (5.12 MB each)
    const size_t rule_elems = (size_t)N_RULES * D;
    float* rule_out = (float*)d_ws;
    float* rule_tmp = rule_out + rule_elems;

    // ---- 1) rule_out = R2V @ x -------------------------------------------
    {
        int n4 = (int)(rule_elems / 4);
        Kongming_zero4<<<(n4 + 255) / 256, 256, 0, stream>>>((float4*)rule_out, n4);
        Kongming_coo_scatter<<<scatter_blocks(nnz_r2v), 256, 0, stream>>>(
            r2v_rows, r2v_cols, r2v_vals, x, rule_out, nnz_r2v);
    }

    // ---- 2) rule_out = R2R_i @ rule_out + rule_out, i = 0..step-1 --------
    // Snapshot old values into rule_tmp, then atomically accumulate
    // R2R_i @ snapshot into rule_out (which already holds the old values).
    for (int i = 0; i < step; ++i) {
        hipMemcpyAsync(rule_tmp, rule_out, rule_elems * sizeof(float),
                       hipMemcpyDeviceToDevice, stream);
        Kongming_coo_scatter<<<scatter_blocks(nnz_r2r), 256, 0, stream>>>(
            r2r_rows + (size_t)i * nnz_r2r,
            r2r_cols + (size_t)i * nnz_r2r,
            r2r_vals + (size_t)i * nnz_r2r,
            rule_tmp, rule_out, nnz_r2r);
    }

    // ---- 3) out = V2R @ rule_out + V2V @ x -------------------------------
    {
        int n4 = ((int)N_NODES * D) / 4;
        Kongming_zero4<<<(n4 + 255) / 256, 256, 0, stream>>>((float4*)out, n4);
        Kongming_coo_scatter<<<scatter_blocks(nnz_v2r), 256, 0, stream>>>(
            v2r_rows, v2r_cols, v2r_vals, rule_out, out, nnz_v2r);
        Kongming_coo_scatter<<<scatter_blocks(nnz_v2v), 256, 0, stream>>>(
            v2v_rows, v2v_cols, v2v_vals, x, out, nnz_v2v);
    }
}